// OrdinalCrossEntropyLoss_69518340653473
// MI455X (gfx1250) — compile-verified
//
#include <hip/hip_runtime.h>
#include <stdint.h>

// OrdinalCrossEntropyLoss: out = sum_n sum_{i<t_n} relu(logits[n,i] - logits[n,t_n])
// N = 262144 rows, C = 512 classes, f32. Pure streaming reduction (512 MB read,
// ~0.4 GFLOP) -> HBM-bound (~22 us @ 23.3 TB/s). Use CDNA5 TDM (tensor_load_to_lds)
// to DMA 64KB row-tiles into LDS, double-buffered, compute from LDS.

#define C_CLASSES 512
#define TILE_ROWS 32
#define THREADS   256
#define WAVES     (THREADS / 32)
#define ROWS_PER_WAVE (TILE_ROWS / WAVES)   // 4
#define NUM_BLOCKS 2048

typedef unsigned int v4u __attribute__((ext_vector_type(4)));
typedef int          v8i __attribute__((ext_vector_type(8)));
typedef int          v4i __attribute__((ext_vector_type(4)));

// Build a TDM descriptor for a 2D tile: TILE_ROWS rows x C_CLASSES f32,
// contiguous in memory starting at gptr, written contiguously at LDS byte
// offset lds_byte_off. Issue the DMA (wave-level; tracked by TENSORcnt).
__device__ __forceinline__ void tdm_load_tile(const float* gptr, unsigned lds_byte_off) {
    uint64_t ga = (uint64_t)(uintptr_t)gptr;

    v4u g0;
    g0[0] = 1u;                                   // count=1 (valid user D#), gather off
    g0[1] = lds_byte_off;                         // lds_addr [63:32]
    g0[2] = (unsigned)(ga & 0xFFFFFFFFu);         // global_addr [95:64]
    g0[3] = (unsigned)((ga >> 32) & 0x1FFFFFFu)   // global_addr [120:96]
          | (2u << 30);                           // type=2 ("image") [127:126]

    v8i g1;
    g1[0] = (int)(2u << 16);                      // workgroup_mask=0, data_size=2 (4B)
    g1[1] = (int)((unsigned)(C_CLASSES & 0xFFFF) << 16);   // tensor_dim0 lo16 @ [63:48]
    g1[2] = (int)(((unsigned)(TILE_ROWS & 0xFFFF) << 16)   // tensor_dim1 lo16 @ [95:80]
          | ((C_CLASSES >> 16) & 0xFFFF));                 // tensor_dim0 hi16 @ [79:64]
    g1[3] = (int)(((unsigned)(C_CLASSES & 0xFFFF) << 16)   // tile_dim0 @ [127:112]
          | ((TILE_ROWS >> 16) & 0xFFFF));                 // tensor_dim1 hi16 @ [111:96]
    g1[4] = TILE_ROWS;                            // tile_dim1 @ [143:128]; tile_dim2=0
    g1[5] = C_CLASSES;                            // tensor_dim0_stride lo32 @ [191:160]
    g1[6] = 0;                                    // stride hi / dim1_stride lo (unused, 2D)
    g1[7] = 0;

    v4i z4 = {0, 0, 0, 0};                        // groups 2/3 unused (2D tile)
    v8i z8 = {0, 0, 0, 0, 0, 0, 0, 0};
    // amdgpu-toolchain (clang-23) 6-arg form:
    // (g0: 4 SGPRs, g1: 8 SGPRs, g2: 4 SGPRs, g3: 4 SGPRs, extra group, cpol)
    __builtin_amdgcn_tensor_load_to_lds(g0, g1, z4, z4, z8, /*cpol*/0);
}

__global__ __launch_bounds__(THREADS)
void ocel_partial_kernel(const float* __restrict__ logits,
                         const long long* __restrict__ targets,
                         float* __restrict__ partial,
                         int n_rows) {
    __shared__ float tile[2][TILE_ROWS * C_CLASSES];   // 2 x 64KB double buffer
    __shared__ float wsum[WAVES];

    const int tid  = threadIdx.x;
    const int wave = tid >> 5;
    const int lane = tid & 31;
    const int tiles_total = n_rows / TILE_ROWS;

    const unsigned lds0 = (unsigned)(uintptr_t)&tile[0][0];
    const unsigned lds1 = (unsigned)(uintptr_t)&tile[1][0];

    float acc = 0.0f;

    const int t0 = (int)blockIdx.x;
    const int stride = (int)gridDim.x;
    const bool lead = (wave == 0);   // one wave issues the DMA (TDM ignores EXEC,
                                     // but the branch keeps other waves from issuing)

    if (lead && t0 < tiles_total) {
        tdm_load_tile(logits + (size_t)t0 * TILE_ROWS * C_CLASSES, lds0);
    }

    int buf = 0;
    for (int t = t0; t < tiles_total; t += stride) {
        const int tn = t + stride;
        if (lead) {
            if (tn < tiles_total) {
                // prefetch next tile into the other buffer, then wait for
                // the oldest (current tile) DMA only
                tdm_load_tile(logits + (size_t)tn * TILE_ROWS * C_CLASSES,
                              buf ? lds0 : lds1);
                __builtin_amdgcn_s_wait_tensorcnt(1);
            } else {
                __builtin_amdgcn_s_wait_tensorcnt(0);
            }
        }
        __syncthreads();   // publish current buffer to all waves

        const float* tb = &tile[buf][0];
        const int row_base = t * TILE_ROWS;

        #pragma unroll
        for (int r = 0; r < ROWS_PER_WAVE; ++r) {
            const int rl  = wave * ROWS_PER_WAVE + r;     // row within tile
            const int row = row_base + rl;
            const int tgt = (int)targets[row];            // wave-uniform
            const float* rowp = tb + rl * C_CLASSES;
            const float tl = rowp[tgt];                   // LDS broadcast read

            #pragma unroll
            for (int chunk = 0; chunk < 4; ++chunk) {
                const int col = lane * 4 + chunk * 128;   // 16B-aligned, conflict-free
                const float4 v = *(const float4*)(rowp + col);
                float d;
                d = v.x - tl; acc += (col + 0 < tgt) ? fmaxf(d, 0.0f) : 0.0f;
                d = v.y - tl; acc += (col + 1 < tgt) ? fmaxf(d, 0.0f) : 0.0f;
                d = v.z - tl; acc += (col + 2 < tgt) ? fmaxf(d, 0.0f) : 0.0f;
                d = v.w - tl; acc += (col + 3 < tgt) ? fmaxf(d, 0.0f) : 0.0f;
            }
        }
        __syncthreads();   // all reads of this buffer done before TDM overwrites it
        buf ^= 1;
    }

    // wave32 reduction (deterministic shuffle tree)
    #pragma unroll
    for (int off = 16; off > 0; off >>= 1)
        acc += __shfl_down(acc, off, 32);
    if (lane == 0) wsum[wave] = acc;
    __syncthreads();

    if (tid == 0) {
        float s = 0.0f;
        #pragma unroll
        for (int w = 0; w < WAVES; ++w) s += wsum[w];   // fixed order
        partial[blockIdx.x] = s;
    }
}

__global__ __launch_bounds__(256)
void ocel_reduce_kernel(const float* __restrict__ partial,
                        float* __restrict__ out, int n) {
    __shared__ float sm[256];
    float s = 0.0f;
    for (int i = (int)threadIdx.x; i < n; i += 256) s += partial[i];  // fixed order
    sm[threadIdx.x] = s;
    __syncthreads();
    #pragma unroll
    for (int off = 128; off > 0; off >>= 1) {
        if ((int)threadIdx.x < off) sm[threadIdx.x] += sm[threadIdx.x + off];
        __syncthreads();
    }
    if (threadIdx.x == 0) out[0] = sm[0];
}

extern "C" void kernel_launch(void* const* d_in, const int* in_sizes, int n_in,
                              void* d_out, int out_size, void* d_ws, size_t ws_size,
                              hipStream_t stream) {
    const float*     logits  = (const float*)d_in[0];
    const long long* targets = (const long long*)d_in[1];   // int64 targets
    float* out     = (float*)d_out;
    float* partial = (float*)d_ws;                          // NUM_BLOCKS floats

    const int n_rows = in_sizes[1];                         // N = 262144

    ocel_partial_kernel<<<NUM_BLOCKS, THREADS, 0, stream>>>(logits, targets, partial, n_rows);
    ocel_reduce_kernel<<<1, 256, 0, stream>>>(partial, out, NUM_BLOCKS);
}